// Attention_24936580120926
// MI455X (gfx1250) — compile-verified
//
#include <hip/hip_runtime.h>
#include <cstdint>
#include <cstddef>

#define B_   2
#define T_   2048
#define E_   1024
#define H_   16
#define HD_  64
#define DI_  64
#define G3_  192
#define ROWS (B_*T_)   // 4096

typedef __attribute__((ext_vector_type(2))) float v2f;
typedef __attribute__((ext_vector_type(8))) float v8f;

// ---------------------------------------------------------------------------
// CDNA5 fp32 WMMA: D(16x16) = A(16x4) * B(4x16) + C
// A frag: lane l -> row M = l%16, k = 2*(l/16) + {0,1}
// B frag: lane l -> col N = l%16, k = 2*(l/16) + {0,1}
// C/D:    vgpr i, lanes 0-15 -> M=i, lanes 16-31 -> M=i+8; N = l%16
// ---------------------------------------------------------------------------
__device__ __forceinline__ v8f wmma4(v2f a, v2f b, v8f c) {
  return __builtin_amdgcn_wmma_f32_16x16x4_f32(false, a, false, b, (short)0, c,
                                               false, false);
}

// gfx1250 async global->LDS copy (16B per lane), tracked by ASYNCcnt.
__device__ __forceinline__ void async_ld_f4(const float* g, const float* lds) {
  uint32_t off = (uint32_t)(uintptr_t)lds;
  asm volatile("global_load_async_to_lds_b128 %0, %1, off"
               :: "v"(off), "v"(g) : "memory");
}
__device__ __forceinline__ void async_wait0() {
  asm volatile("s_wait_asynccnt 0" ::: "memory");
}

// ---------------------------------------------------------------------------
// Kernel 1: pooling + silu*v_emb gather + xw = xpos @ wih^T (q and k)
// ---------------------------------------------------------------------------
__global__ __launch_bounds__(256)
void prep_kernel(const float* __restrict__ x, const int* __restrict__ tok,
                 const float* __restrict__ v_emb,
                 const float* __restrict__ wih_q, const float* __restrict__ wih_k,
                 float* __restrict__ xpos, float* __restrict__ xproj,
                 float* __restrict__ vbuf,
                 float* __restrict__ xw_q, float* __restrict__ xw_k)
{
  const int row = blockIdx.x, tid = threadIdx.x;
  __shared__ float xr[E_];
  __shared__ float xp[DI_];
  for (int i = tid; i < E_; i += 256) xr[i] = x[(size_t)row * E_ + i];
  __syncthreads();

  if (tid < DI_) {                       // block-mean of 16 (exact pool 1024->64)
    float s = 0.f;
    for (int u = 0; u < 16; ++u) s += xr[tid * 16 + u];
    s *= (1.f / 16.f);
    xp[tid] = s;
    xpos[(size_t)row * DI_ + tid] = s;
  }
  for (int i = tid; i < 960; i += 256) { // pool 1024->960 (1-2 elem windows)
    int s0 = (i * 16) / 15;
    int e0 = ((i + 1) * 16 + 14) / 15;
    float s = 0.f;
    for (int u = s0; u < e0; ++u) s += xr[u];
    xproj[(size_t)row * 960 + i] = s / (float)(e0 - s0);
  }
  int tk = tok[row];
  for (int c = tid; c < E_; c += 256) {  // v = silu(x) * v_emb[tok]
    float xv = xr[c];
    float sil = xv / (1.f + __expf(-xv));
    vbuf[(size_t)row * E_ + c] = sil * v_emb[(size_t)tk * E_ + c];
  }
  __syncthreads();
  if (tid < G3_) {                       // xw[g] = sum_d xpos[d]*wih[g,d]
    float aq = 0.f, ak = 0.f;
    for (int d = 0; d < DI_; ++d) {
      float xd = xp[d];
      aq += wih_q[tid * DI_ + d] * xd;
      ak += wih_k[tid * DI_ + d] * xd;
    }
    xw_q[(size_t)row * G3_ + tid] = aq;
    xw_k[(size_t)row * G3_ + tid] = ak;
  }
}

// ---------------------------------------------------------------------------
// Kernel 2: GRU scan. 4 blocks = {q,k} x {batch0,1}; whh rows in registers,
// h/hw in LDS; latency chain of 2048 steps.
// ---------------------------------------------------------------------------
__global__ __launch_bounds__(192)
void gru_kernel(const float* __restrict__ xw_qg, const float* __restrict__ xw_kg,
                const float* __restrict__ whh_q, const float* __restrict__ whh_k,
                float* __restrict__ pos_q, float* __restrict__ pos_k)
{
  const int which = blockIdx.x >> 1;
  const int b = blockIdx.x & 1;
  const float* xw  = which ? xw_kg : xw_qg;
  const float* whh = which ? whh_k : whh_q;
  float* pos       = which ? pos_k : pos_q;
  const int g = threadIdx.x;             // 0..191 (one gate-row per thread)

  float w[DI_];
  for (int d = 0; d < DI_; ++d) w[d] = whh[g * DI_ + d];

  __shared__ float h[DI_];
  __shared__ float hw[G3_];
  if (g < DI_) h[g] = 0.f;
  __syncthreads();

  for (int t = 0; t < T_; ++t) {
    float acc = 0.f;
    for (int d = 0; d < DI_; ++d) acc += w[d] * h[d];
    hw[g] = acc;
    __syncthreads();
    if (g < DI_) {
      const float* xr = xw + (size_t)(b * T_ + t) * G3_;
      float r = 1.f / (1.f + __expf(-(xr[g]       + hw[g])));
      float z = 1.f / (1.f + __expf(-(xr[64 + g]  + hw[64 + g])));
      float n = tanhf(xr[128 + g] + r * hw[128 + g]);
      float hn = (1.f - z) * n + z * h[g];
      h[g] = hn;
      pos[(size_t)(b * T_ + t) * DI_ + g] = hn;
    }
    __syncthreads();
  }
}

// ---------------------------------------------------------------------------
// Kernel 3: generic fp32 WMMA GEMM  C[M,N] = [A0|A1][M,K] * W[N,K]^T (+R)(+bias)
// A is split at k0 (k0 multiple of 64). Block tile 128x128, K-tile 64,
// 8 waves each 32x64. Tiles loaded with async global->LDS.
// ---------------------------------------------------------------------------
__global__ __launch_bounds__(256)
void gemm_wmma_kernel(const float* __restrict__ A0, int lda0, int k0,
                      const float* __restrict__ A1, int lda1,
                      const float* __restrict__ W,
                      const float* __restrict__ R,
                      const float* __restrict__ bias,
                      float* __restrict__ C, int N, int K)
{
  __shared__ float As[128 * 64];
  __shared__ float Bs[128 * 64];
  const int tid = threadIdx.x;
  const int lane = tid & 31;
  const int wv = tid >> 5;
  const int wy = wv >> 1, wx = wv & 1;   // 4x2 wave grid -> 32x64 per wave
  const int bm = blockIdx.y * 128;
  const int bn = blockIdx.x * 128;

  v8f acc[2][4] = {};

  const int nkt = K >> 6;
  for (int kt = 0; kt < nkt; ++kt) {
    const int kb = kt << 6;
    __syncthreads();                     // protect LDS reuse
    for (int i = 0; i < 8; ++i) {        // A tile 128x64 : 8 float4/thread
      int idx = tid + (i << 8);
      int r = idx >> 4, c4 = (idx & 15) << 2;
      const float* src = (kb < k0)
          ? A0 + (size_t)(bm + r) * lda0 + kb + c4
          : A1 + (size_t)(bm + r) * lda1 + (kb - k0) + c4;
      async_ld_f4(src, &As[r * 64 + c4]);
    }
    for (int i = 0; i < 8; ++i) {        // W tile 128x64
      int idx = tid + (i << 8);
      int r = idx >> 4, c4 = (idx & 15) << 2;
      async_ld_f4(W + (size_t)(bn + r) * K + kb + c4, &Bs[r * 64 + c4]);
    }
    async_wait0();
    __syncthreads();

    for (int kk = 0; kk < 64; kk += 4) {
      const int ksub = kk + ((lane >> 4) << 1);
      v2f af[2], bf[4];
      for (int im = 0; im < 2; ++im)
        af[im] = *(const v2f*)&As[(wy * 32 + im * 16 + (lane & 15)) * 64 + ksub];
      for (int jn = 0; jn < 4; ++jn)
        bf[jn] = *(const v2f*)&Bs[(wx * 64 + jn * 16 + (lane & 15)) * 64 + ksub];
      for (int im = 0; im < 2; ++im)
        for (int jn = 0; jn < 4; ++jn)
          acc[im][jn] = wmma4(af[im], bf[jn], acc[im][jn]);
    }
  }

  const int rhalf = (lane >> 4) << 3;    // 0 or 8
  const int ncol = lane & 15;
  for (int im = 0; im < 2; ++im)
    for (int jn = 0; jn < 4; ++jn)
      for (int i = 0; i < 8; ++i) {
        int row = bm + wy * 32 + im * 16 + rhalf + i;
        int col = bn + wx * 64 + jn * 16 + ncol;
        float v = acc[im][jn][i];
        if (R)    v += R[(size_t)row * N + col];
        if (bias) v += bias[col];
        C[(size_t)row * N + col] = v;
      }
}

// ---------------------------------------------------------------------------
// Kernel 4: row LayerNorm, in place. One block per row.
// ---------------------------------------------------------------------------
__global__ __launch_bounds__(256)
void ln_kernel(float* __restrict__ buf, const float* __restrict__ w,
               const float* __restrict__ b)
{
  const int row = blockIdx.x, tid = threadIdx.x;
  const float* p = buf + (size_t)row * E_;
  float vals[4];
  float s = 0.f, s2 = 0.f;
  for (int i = 0; i < 4; ++i) {
    float v = p[tid + i * 256];
    vals[i] = v; s += v; s2 += v * v;
  }
  __shared__ float rs[256], rs2[256];
  rs[tid] = s; rs2[tid] = s2;
  __syncthreads();
  for (int off = 128; off > 0; off >>= 1) {
    if (tid < off) { rs[tid] += rs[tid + off]; rs2[tid] += rs2[tid + off]; }
    __syncthreads();
  }
  float mu = rs[0] * (1.f / E_);
  float var = rs2[0] * (1.f / E_) - mu * mu;
  float inv = rsqrtf(var + 1e-5f);
  for (int i = 0; i < 4; ++i) {
    int c = tid + i * 256;
    buf[(size_t)row * E_ + c] = (vals[i] - mu) * inv * w[c] + b[c];
  }
}

// ---------------------------------------------------------------------------
// Kernel 5: flash attention per (b,h,q-tile of 128). Key tile 64.
// Scale 2/sqrt(64)=0.25 folded into Q. Causal mask -> -1e30.
// Per wave: 16 query rows; S via WMMA; online softmax with 16-lane shuffles;
// P staged through LDS, PV via WMMA.
// ---------------------------------------------------------------------------
__global__ __launch_bounds__(256)
void flash_kernel(const float* __restrict__ q, const float* __restrict__ k,
                  const float* __restrict__ v, float* __restrict__ o)
{
  __shared__ float Qs[128 * 64];
  __shared__ float Ks[64 * 64];
  __shared__ float Vs[64 * 64];
  __shared__ float Ps[128 * 64];
  const int tid = threadIdx.x, lane = tid & 31, wv = tid >> 5;
  const int qt = blockIdx.x, h = blockIdx.y, b = blockIdx.z;
  const int qbase = qt * 128;
  const int rb = wv * 16;
  const int rhalf = (lane >> 4) << 3;
  const int ncol = lane & 15;

  for (int idx = tid; idx < 128 * 64; idx += 256) {
    int r = idx >> 6, c = idx & 63;
    Qs[idx] = q[(size_t)(b * T_ + qbase + r) * E_ + h * HD_ + c] * 0.25f;
  }

  float m_i[8], l_i[8];
  v8f oacc[4] = {};
  for (int i = 0; i < 8; ++i) { m_i[i] = -1e30f; l_i[i] = 0.f; }

  const int jtmax = 2 * qt + 1;
  for (int jt = 0; jt <= jtmax; ++jt) {
    const int jbase = jt * 64;
    __syncthreads();                     // previous tile fully consumed
    for (int i = 0; i < 4; ++i) {        // K,V tiles 64x64 each
      int idx = tid + (i << 8);
      int r = idx >> 4, c4 = (idx & 15) << 2;
      size_t goff = (size_t)(b * T_ + jbase + r) * E_ + h * HD_ + c4;
      async_ld_f4(k + goff, &Ks[r * 64 + c4]);
      async_ld_f4(v + goff, &Vs[r * 64 + c4]);
    }
    async_wait0();
    __syncthreads();

    // S = Q * K^T  (16 x 64 per wave)
    v8f sacc[4] = {};
    for (int kk = 0; kk < 64; kk += 4) {
      const int ksub = kk + ((lane >> 4) << 1);
      v2f a = *(const v2f*)&Qs[(rb + (lane & 15)) * 64 + ksub];
      for (int jn = 0; jn < 4; ++jn) {
        v2f bb = *(const v2f*)&Ks[(jn * 16 + (lane & 15)) * 64 + ksub];
        sacc[jn] = wmma4(a, bb, sacc[jn]);
      }
    }

    // causal mask + online softmax
    float p[4][8];
    for (int i = 0; i < 8; ++i) {
      int rowg = qbase + rb + rhalf + i;
      float rmax = -1e30f;
      for (int jn = 0; jn < 4; ++jn) {
        int colg = jbase + jn * 16 + ncol;
        float s = (colg > rowg) ? -1e30f : sacc[jn][i];
        p[jn][i] = s;
        rmax = fmaxf(rmax, s);
      }
      for (int m = 1; m < 16; m <<= 1) rmax = fmaxf(rmax, __shfl_xor(rmax, m, 32));
      float mnew = fmaxf(m_i[i], rmax);
      float alpha = __expf(m_i[i] - mnew);
      float rsum = 0.f;
      for (int jn = 0; jn < 4; ++jn) {
        float pe = __expf(p[jn][i] - mnew);
        p[jn][i] = pe;
        rsum += pe;
      }
      for (int m = 1; m < 16; m <<= 1) rsum += __shfl_xor(rsum, m, 32);
      l_i[i] = l_i[i] * alpha + rsum;
      m_i[i] = mnew;
      for (int dt = 0; dt < 4; ++dt) oacc[dt][i] = oacc[dt][i] * alpha;
    }

    // stage P (C-layout) -> LDS so it can be read back in A-layout
    for (int jn = 0; jn < 4; ++jn)
      for (int i = 0; i < 8; ++i)
        Ps[(rb + rhalf + i) * 64 + jn * 16 + ncol] = p[jn][i];
    // same-wave LDS ops are in-order; no barrier needed before own reads

    // O += P * V
    for (int kk = 0; kk < 64; kk += 4) {
      const int ksub = kk + ((lane >> 4) << 1);
      v2f a = *(const v2f*)&Ps[(rb + (lane & 15)) * 64 + ksub];
      for (int dt = 0; dt < 4; ++dt) {
        v2f bb;
        bb.x = Vs[(ksub + 0) * 64 + dt * 16 + ncol];
        bb.y = Vs[(ksub + 1) * 64 + dt * 16 + ncol];
        oacc[dt] = wmma4(a, bb, oacc[dt]);
      }
    }
  }

  for (int dt = 0; dt < 4; ++dt)
    for (int i = 0; i < 8; ++i) {
      int r = rb + rhalf + i;
      o[(size_t)(b * T_ + qbase + r) * E_ + h * HD_ + dt * 16 + ncol] =
          oacc[dt][i] / l_i[i];
    }
}

// ---------------------------------------------------------------------------
extern "C" void kernel_launch(void* const* d_in, const int* in_sizes, int n_in,
                              void* d_out, int out_size, void* d_ws, size_t ws_size,
                              hipStream_t stream)
{
  const float* x     = (const float*)d_in[0];
  const int*   tok   = (const int*)d_in[1];
  const float* q_wih = (const float*)d_in[2];
  const float* q_whh = (const float*)d_in[3];
  const float* q_lin = (const float*)d_in[4];
  const float* k_wih = (const float*)d_in[5];
  const float* k_whh = (const float*)d_in[6];
  const float* k_lin = (const float*)d_in[7];
  const float* v_emb = (const float*)d_in[8];
  const float* q_lnw = (const float*)d_in[9];
  const float* q_lnb = (const float*)d_in[10];
  const float* k_lnw = (const float*)d_in[11];
  const float* k_lnb = (const float*)d_in[12];
  const float* out_w = (const float*)d_in[13];
  const float* out_b = (const float*)d_in[14];
  float* out = (float*)d_out;

  float* ws    = (float*)d_ws;
  float* xpos  = ws;                      // 4096*64
  float* xproj = xpos  + (size_t)ROWS * DI_;   // 4096*960
  float* xw_q  = xproj + (size_t)ROWS * 960;   // 4096*192
  float* xw_k  = xw_q  + (size_t)ROWS * G3_;
  float* pos_q = xw_k  + (size_t)ROWS * G3_;   // 4096*64
  float* pos_k = pos_q + (size_t)ROWS * DI_;
  float* qb    = pos_k + (size_t)ROWS * DI_;   // 4096*1024
  float* kb    = qb    + (size_t)ROWS * E_;
  float* vb    = kb    + (size_t)ROWS * E_;
  float* ab    = vb    + (size_t)ROWS * E_;

  prep_kernel<<<ROWS, 256, 0, stream>>>(x, tok, v_emb, q_wih, k_wih,
                                        xpos, xproj, vb, xw_q, xw_k);
  gru_kernel<<<4, 192, 0, stream>>>(xw_q, xw_k, q_whh, k_whh, pos_q, pos_k);

  dim3 gg(E_ / 128, ROWS / 128);
  // q/k pre-LN: [pos | xproj] @ lin_w^T + x
  gemm_wmma_kernel<<<gg, 256, 0, stream>>>(pos_q, DI_, 64, xproj, 960,
                                           q_lin, x, nullptr, qb, E_, E_);
  gemm_wmma_kernel<<<gg, 256, 0, stream>>>(pos_k, DI_, 64, xproj, 960,
                                           k_lin, x, nullptr, kb, E_, E_);
  ln_kernel<<<ROWS, 256, 0, stream>>>(qb, q_lnw, q_lnb);
  ln_kernel<<<ROWS, 256, 0, stream>>>(kb, k_lnw, k_lnb);

  flash_kernel<<<dim3(T_ / 128, H_, B_), 256, 0, stream>>>(qb, kb, vb, ab);

  // out = attn @ out_w^T + out_b
  gemm_wmma_kernel<<<gg, 256, 0, stream>>>(ab, E_, E_, ab, E_,
                                           out_w, nullptr, out_b, out, E_, E_);
}